// TopKRouter_8297876816194
// MI455X (gfx1250) — compile-verified
//
#include <hip/hip_runtime.h>
#include <math.h>

typedef __attribute__((ext_vector_type(2))) float v2f;
typedef __attribute__((ext_vector_type(8))) float v8f;

// Problem constants (match reference setup_inputs)
constexpr int B_ = 4;
constexpr int S_ = 4096;
constexpr int D_ = 2048;
constexpr int E_ = 64;
constexpr int NT = B_ * S_;            // 16384 tokens
constexpr int NTILE = E_ / 16;         // 4 N-tiles of 16 experts
constexpr int WAVES_PER_BLOCK = 4;     // K split 4 ways
constexpr int BLOCK = WAVES_PER_BLOCK * 32;
constexpr int KSLICE = D_ / WAVES_PER_BLOCK;   // 512 K per wave

__device__ __forceinline__ bool better(float v, int e, float bv, int be) {
    // strictly better candidate; ties broken by lower expert index (jax top_k)
    return (v > bv) || (v == bv && e < be);
}

__global__ __launch_bounds__(BLOCK)
void topk_router_wmma(const float* __restrict__ x,
                      const float* __restrict__ Wr,
                      float* __restrict__ gate,
                      int*   __restrict__ eidx,
                      float* __restrict__ probs)
{
    __shared__ float lds[WAVES_PER_BLOCK * 16 * E_];   // 16KB partial logits

    const int tid     = threadIdx.x;
    const int lane    = tid & 31;
    const int lane16  = lane & 15;         // A row M / B col N within tile
    const int khalf   = lane >> 4;         // 0 -> K={0,1}, 1 -> K={2,3}
    const int wib     = tid >> 5;          // wave in block: K-slice id
    const int m0      = blockIdx.x * 16;   // 16 tokens per block
    const int k0      = wib * KSLICE;

    // ---- partial GEMM: logits[m0..m0+15][0..63] over K in [k0, k0+512) ----
    v8f acc[NTILE] = {};

    // A (16x4 f32): lane16 = row M; khalf selects K pair {0,1}/{2,3}
    const float* aptr = x + (size_t)(m0 + lane16) * D_ + k0 + 2 * khalf;
    // B (4x16 f32): lane16 = col N (expert); B[k][n] = Wr[n][k]
    const float* bptr = Wr + (size_t)lane16 * D_ + k0 + 2 * khalf;

#pragma unroll 4
    for (int k = 0; k < KSLICE; k += 4) {
        // x is streamed once -> non-temporal so it doesn't evict L2-hot Wr
        v2f a = __builtin_nontemporal_load((const v2f*)(aptr + k));
#pragma unroll
        for (int t = 0; t < NTILE; ++t) {
            v2f b = *(const v2f*)(bptr + (size_t)t * 16 * D_ + k);
            acc[t] = __builtin_amdgcn_wmma_f32_16x16x4_f32(
                false, a, false, b, (short)0, acc[t], false, false);
        }
    }

    // ---- stash partials: C layout VGPR r = rows M=r / M=r+8, N = lane16 ----
    float* my = lds + wib * 16 * E_;
#pragma unroll
    for (int t = 0; t < NTILE; ++t)
#pragma unroll
        for (int r = 0; r < 8; ++r)
            my[(r + khalf * 8) * E_ + t * 16 + lane16] = acc[t][r];

    __syncthreads();

    // ---- fused epilogue: 8 threads per token, 8 experts per thread ----
    const int token = tid >> 3;            // 0..15
    const int sub   = tid & 7;             // 0..7 (xor 1,2,4 stays in-group)
    const int gtok  = m0 + token;

    float v[8];
#pragma unroll
    for (int j = 0; j < 8; ++j) {
        const int e = sub * 8 + j;
        float s0 = lds[0 * 16 * E_ + token * E_ + e];
        float s1 = lds[1 * 16 * E_ + token * E_ + e];
        float s2 = lds[2 * 16 * E_ + token * E_ + e];
        float s3 = lds[3 * 16 * E_ + token * E_ + e];
        v[j] = (s0 + s1) + (s2 + s3);
    }

    // softmax max
    float mx = v[0];
#pragma unroll
    for (int j = 1; j < 8; ++j) mx = fmaxf(mx, v[j]);
#pragma unroll
    for (int ofs = 1; ofs <= 4; ofs <<= 1)
        mx = fmaxf(mx, __shfl_xor(mx, ofs, 32));

    // exp + sum
    float p[8], s = 0.0f;
#pragma unroll
    for (int j = 0; j < 8; ++j) { p[j] = __expf(v[j] - mx); s += p[j]; }
#pragma unroll
    for (int ofs = 1; ofs <= 4; ofs <<= 1)
        s += __shfl_xor(s, ofs, 32);
    const float inv = 1.0f / s;

    float pr[8];
#pragma unroll
    for (int j = 0; j < 8; ++j) {
        pr[j] = p[j] * inv;
        probs[(size_t)gtok * E_ + sub * 8 + j] = pr[j];
    }

    // local top-2 among this thread's 8 experts
    float b1v = -INFINITY, b2v = -INFINITY;
    int   b1e = E_,        b2e = E_;
#pragma unroll
    for (int j = 0; j < 8; ++j) {
        const float pv = pr[j];
        const int   pe = sub * 8 + j;
        if (better(pv, pe, b1v, b1e)) {
            b2v = b1v; b2e = b1e; b1v = pv; b1e = pe;
        } else if (better(pv, pe, b2v, b2e)) {
            b2v = pv; b2e = pe;
        }
    }

    // butterfly merge of sorted top-2 pairs across the 8-thread group
#pragma unroll
    for (int ofs = 1; ofs <= 4; ofs <<= 1) {
        float o1v = __shfl_xor(b1v, ofs, 32);
        int   o1e = __shfl_xor(b1e, ofs, 32);
        float o2v = __shfl_xor(b2v, ofs, 32);
        int   o2e = __shfl_xor(b2e, ofs, 32);
        if (better(o1v, o1e, b1v, b1e)) {
            float c2v; int c2e;
            if (better(b1v, b1e, o2v, o2e)) { c2v = b1v; c2e = b1e; }
            else                            { c2v = o2v; c2e = o2e; }
            b1v = o1v; b1e = o1e; b2v = c2v; b2e = c2e;
        } else if (better(o1v, o1e, b2v, b2e)) {
            b2v = o1v; b2e = o1e;
        }
    }

    if (sub == 0) {
        const float den = b1v + b2v;
        gate[gtok * 2 + 0] = b1v / den;
        gate[gtok * 2 + 1] = b2v / den;
        eidx[gtok * 2 + 0] = b1e;
        eidx[gtok * 2 + 1] = b2e;
    }
}

extern "C" void kernel_launch(void* const* d_in, const int* in_sizes, int n_in,
                              void* d_out, int out_size, void* d_ws, size_t ws_size,
                              hipStream_t stream) {
    const float* x  = (const float*)d_in[0];   // [B,S,D] f32
    const float* Wr = (const float*)d_in[1];   // [E,D]   f32

    // d_out = [gate (NT*2 f32)][expert_idx (NT*2 i32)][router_probs (NT*64 f32)]
    float* out   = (float*)d_out;
    float* gate  = out;
    int*   eidx  = (int*)(out + 2 * NT);
    float* probs = out + 4 * NT;

    const int grid = NT / 16;                  // 1024 blocks x 128 threads = 4096 waves
    topk_router_wmma<<<grid, BLOCK, 0, stream>>>(x, Wr, gate, eidx, probs);
}